// graph_encoder_52888227283046
// MI455X (gfx1250) — compile-verified
//
#include <hip/hip_runtime.h>
#include <hip/hip_bf16.h>

// ---------------------------------------------------------------------------
// MI455X / gfx1250 fused graph-attention encoder.
// GEMMs on V_WMMA_F32_16X16X4_F32 (exact f32, matrix pipe).
// Round 3: async global->LDS double-buffered K/V tile staging in attention
// (GLOBAL_LOAD_ASYNC_TO_LDS_B128 + s_wait_asynccnt), 16B-aligned padded tiles.
// ---------------------------------------------------------------------------

typedef __attribute__((ext_vector_type(2))) float v2f;
typedef __attribute__((ext_vector_type(8))) float v8f;

static constexpr int kB   = 128;
static constexpr int kENV = 512;

__device__ __forceinline__ v8f wmma4(v2f a, v2f b, v8f c) {
  return __builtin_amdgcn_wmma_f32_16x16x4_f32(
      false, a, false, b, (short)0, c, false, false);
}

__device__ __forceinline__ v2f ld2(const float* p) { return *(const v2f*)p; }

// Issue an async copy of a 64 x D f32 tile (row stride DS floats in LDS,
// DS*4 must be a multiple of 16) from global to LDS. Tracked by ASYNCcnt;
// (64*D/4)/128 instructions per thread.
template <int D, int DS>
__device__ __forceinline__ void issue_tile_async(float* dstLds,
                                                 const float* srcGlobal) {
  constexpr int CPR = D / 4;                    // b128 chunks per row
  const int tid = threadIdx.x;
#pragma unroll
  for (int n = 0; n < (64 * CPR) / 128; ++n) {
    const int idx = n * 128 + tid;
    const int r = idx / CPR, c4 = (idx - r * CPR) * 4;
    unsigned lds = (unsigned)(uintptr_t)(dstLds + r * DS + c4);
    const float* gp = srcGlobal + (size_t)r * D + c4;
    asm volatile("global_load_async_to_lds_b128 %0, %1, off"
                 :: "v"(lds), "v"(gp) : "memory");
  }
}

template <int N>
__device__ __forceinline__ void wait_async() {
  asm volatile("s_wait_asynccnt %0" :: "i"(N) : "memory");
}

// ---------------------------------------------------------------------------
// Two-layer MLP: Y = relu(X @ W1 + b1) @ W2 + b2
// Weights staged TRANSPOSED in LDS: sWt[n*KS + k] = W[k][n].
// ---------------------------------------------------------------------------
template <int DIN, int DINP, int DH, int DOUT>
__launch_bounds__(128)
__global__ void mlp2_kernel(const float* __restrict__ X,
                            const float* __restrict__ W1,
                            const float* __restrict__ b1,
                            const float* __restrict__ W2,
                            const float* __restrict__ b2,
                            float* __restrict__ Y) {
  constexpr int SXS = DINP + 2;
  constexpr int K1S = DINP + 2;
  constexpr int K2S = DH + 2;
  constexpr int SHS = DH + 2;
  constexpr int W1SZ = DH * K1S;
  constexpr int W2SZ = DOUT * K2S;
  constexpr int WSZ  = (W1SZ > W2SZ) ? W1SZ : W2SZ;

  extern __shared__ float smem[];
  float* sX = smem;                 // 64 * SXS
  float* sW = sX + 64 * SXS;        // WSZ  (W1ᵀ then W2ᵀ)
  float* sH = sW + WSZ;             // 64 * SHS

  const int tid  = threadIdx.x;
  const int lane = tid & 31;
  const int wave = tid >> 5;
  const int b    = blockIdx.y;
  const int row0 = blockIdx.x * 64;

  for (int i = tid; i < 64 * DINP; i += blockDim.x) {
    int r = i / DINP, c = i - r * DINP;
    sX[r * SXS + c] = (c < DIN) ? X[((size_t)b * kENV + row0 + r) * DIN + c] : 0.f;
  }
  for (int i = tid; i < DINP * DH; i += blockDim.x) {
    int r = i / DH, c = i - r * DH;
    sW[c * K1S + r] = (r < DIN) ? W1[(size_t)r * DH + c] : 0.f;
  }
  __syncthreads();

  const int arow = wave * 16;
  const int rsel = lane >> 4;
  const int rlo  = lane & 15;

  {
    constexpr int NT = DH / 16;
    v8f acc[NT];
#pragma unroll
    for (int nt = 0; nt < NT; ++nt) {
      const float bi = b1[nt * 16 + rlo];
#pragma unroll
      for (int i = 0; i < 8; ++i) acc[nt][i] = bi;
    }
#pragma unroll
    for (int k0 = 0; k0 < DINP; k0 += 4) {
      const int ka = k0 + (rsel << 1);
      const v2f a = ld2(&sX[(arow + rlo) * SXS + ka]);
#pragma unroll
      for (int nt = 0; nt < NT; ++nt) {
        const v2f bb = ld2(&sW[(nt * 16 + rlo) * K1S + ka]);
        acc[nt] = wmma4(a, bb, acc[nt]);
      }
    }
#pragma unroll
    for (int nt = 0; nt < NT; ++nt)
#pragma unroll
      for (int i = 0; i < 8; ++i)
        sH[(arow + i + 8 * rsel) * SHS + nt * 16 + rlo] = fmaxf(acc[nt][i], 0.f);
  }
  __syncthreads();

  for (int i = tid; i < DH * DOUT; i += blockDim.x) {
    int r = i / DOUT, c = i - r * DOUT;
    sW[c * K2S + r] = W2[i];
  }
  __syncthreads();

  {
    constexpr int NT = DOUT / 16;
    v8f acc[NT];
#pragma unroll
    for (int nt = 0; nt < NT; ++nt) {
      const float bi = b2[nt * 16 + rlo];
#pragma unroll
      for (int i = 0; i < 8; ++i) acc[nt][i] = bi;
    }
#pragma unroll
    for (int k0 = 0; k0 < DH; k0 += 4) {
      const int ka = k0 + (rsel << 1);
      const v2f a = ld2(&sH[(arow + rlo) * SHS + ka]);
#pragma unroll
      for (int nt = 0; nt < NT; ++nt) {
        const v2f bb = ld2(&sW[(nt * 16 + rlo) * K2S + ka]);
        acc[nt] = wmma4(a, bb, acc[nt]);
      }
    }
#pragma unroll
    for (int nt = 0; nt < NT; ++nt)
#pragma unroll
      for (int i = 0; i < 8; ++i)
        Y[((size_t)b * kENV + row0 + arow + i + 8 * rsel) * DOUT + nt * 16 + rlo] =
            acc[nt][i];
  }
}

// ---------------------------------------------------------------------------
// Attention + LayerNorm with async double-buffered K/V tiles.
// ---------------------------------------------------------------------------
template <int D>
__launch_bounds__(128)
__global__ void attn_ln_kernel(const float* __restrict__ Q,
                               const float* __restrict__ K,
                               const float* __restrict__ V,
                               const float* __restrict__ g,
                               const float* __restrict__ beta,
                               float* __restrict__ Y) {
  constexpr int DS     = D + 4;                 // 16B-aligned padded stride
  constexpr int SSS    = 512 + 2;
  constexpr int CHUNKS = (64 * (D / 4)) / 128;  // async instr per thread per tile

  extern __shared__ float smem[];
  float* sQ  = smem;                  // 64 * DS
  float* sB0 = sQ + 64 * DS;          // tile buffer 0
  float* sB1 = sB0 + 64 * DS;         // tile buffer 1
  float* sS  = sB1 + 64 * DS;         // 64 * SSS

  const int tid  = threadIdx.x;
  const int lane = tid & 31;
  const int wave = tid >> 5;
  const int b    = blockIdx.y;
  const int row0 = blockIdx.x * 64;
  const float scale = rsqrtf((float)D);
  float* buf[2] = { sB0, sB1 };

  // Prologue: async-stage Q tile and K tile 0.
  issue_tile_async<D, DS>(sQ, Q + ((size_t)b * kENV + row0) * D);
  issue_tile_async<D, DS>(sB0, K + ((size_t)b * kENV) * D);

  const int arow = wave * 16;
  const int rsel = lane >> 4;
  const int rlo  = lane & 15;

  // ---- S = (Q @ Kᵀ) * scale, K tiles double-buffered ----
  for (int kt = 0; kt < kENV / 64; ++kt) {
    const int cur = kt & 1, nxt = cur ^ 1;
    if (kt < kENV / 64 - 1) {
      issue_tile_async<D, DS>(buf[nxt],
                              K + ((size_t)b * kENV + (kt + 1) * 64) * D);
      wait_async<CHUNKS>();    // oldest group (tile kt, +Q at kt==0) complete
    } else {
      wait_async<0>();
    }
    __syncthreads();

    v8f acc[4];
#pragma unroll
    for (int nt = 0; nt < 4; ++nt)
#pragma unroll
      for (int i = 0; i < 8; ++i) acc[nt][i] = 0.f;
#pragma unroll
    for (int k0 = 0; k0 < D; k0 += 4) {
      const int ka = k0 + (rsel << 1);
      const v2f a = ld2(&sQ[(arow + rlo) * DS + ka]);
#pragma unroll
      for (int nt = 0; nt < 4; ++nt) {
        const v2f bb = ld2(&buf[cur][(nt * 16 + rlo) * DS + ka]);  // Kᵀ pair
        acc[nt] = wmma4(a, bb, acc[nt]);
      }
    }
#pragma unroll
    for (int nt = 0; nt < 4; ++nt)
#pragma unroll
      for (int i = 0; i < 8; ++i)
        sS[(arow + i + 8 * rsel) * SSS + kt * 64 + nt * 16 + rlo] =
            acc[nt][i] * scale;
    __syncthreads();   // all waves done with buf[cur] before it is re-filled
  }

  // Stage V tile 0 while softmax runs.
  issue_tile_async<D, DS>(buf[0], V + ((size_t)b * kENV) * D);

  // ---- row softmax over 512 keys ----
  if (tid < 64) {
    float* row = sS + tid * SSS;
    float m = -3.4e38f;
    for (int j = 0; j < 512; ++j) m = fmaxf(m, row[j]);
    float s = 0.f;
    for (int j = 0; j < 512; ++j) { float e = __expf(row[j] - m); row[j] = e; s += e; }
    const float inv = 1.f / s;
    for (int j = 0; j < 512; ++j) row[j] *= inv;
  }

  // ---- O = P @ V, V tiles double-buffered ----
  constexpr int NT = D / 16;
  v8f o[NT];
#pragma unroll
  for (int nt = 0; nt < NT; ++nt)
#pragma unroll
    for (int i = 0; i < 8; ++i) o[nt][i] = 0.f;

  for (int vt = 0; vt < kENV / 64; ++vt) {
    const int cur = vt & 1, nxt = cur ^ 1;
    if (vt < kENV / 64 - 1) {
      issue_tile_async<D, DS>(buf[nxt],
                              V + ((size_t)b * kENV + (vt + 1) * 64) * D);
      wait_async<CHUNKS>();
    } else {
      wait_async<0>();
    }
    __syncthreads();   // also orders softmax writes before first PV read

#pragma unroll
    for (int k0 = 0; k0 < 64; k0 += 4) {
      const int ka = k0 + (rsel << 1);
      const v2f a = ld2(&sS[(arow + rlo) * SSS + vt * 64 + ka]);
#pragma unroll
      for (int nt = 0; nt < NT; ++nt) {
        v2f bb;
        bb.x = buf[cur][ka * DS + nt * 16 + rlo];
        bb.y = buf[cur][(ka + 1) * DS + nt * 16 + rlo];
        o[nt] = wmma4(a, bb, o[nt]);
      }
    }
    __syncthreads();
  }

  // ---- spill O, fused LayerNorm, store ----
#pragma unroll
  for (int nt = 0; nt < NT; ++nt)
#pragma unroll
    for (int i = 0; i < 8; ++i)
      sQ[(arow + i + 8 * rsel) * DS + nt * 16 + rlo] = o[nt][i];
  __syncthreads();

  if (tid < 64) {
    const float* row = sQ + tid * DS;
    float m = 0.f;
    for (int d = 0; d < D; ++d) m += row[d];
    m /= (float)D;
    float v = 0.f;
    for (int d = 0; d < D; ++d) { float t = row[d] - m; v += t * t; }
    v /= (float)D;
    const float inv = rsqrtf(v + 1e-5f);
    for (int d = 0; d < D; ++d)
      Y[((size_t)b * kENV + row0 + tid) * D + d] = (row[d] - m) * inv * g[d] + beta[d];
  }
}

// ---------------------------------------------------------------------------
// Aggregator stage 1: h = relu(X @ W1 + b1), logits = h . ws   (D = 128)
// ---------------------------------------------------------------------------
__launch_bounds__(128)
__global__ void agg_h_kernel(const float* __restrict__ X,
                             const float* __restrict__ W1,
                             const float* __restrict__ b1,
                             const float* __restrict__ wsv,
                             float* __restrict__ H,
                             float* __restrict__ logits) {
  constexpr int ST = 130;
  extern __shared__ float smem[];
  float* sX = smem;
  float* sW = sX + 64 * ST;
  float* sH = sW + 128 * ST;

  const int tid  = threadIdx.x;
  const int lane = tid & 31;
  const int wave = tid >> 5;
  const int b    = blockIdx.y;
  const int row0 = blockIdx.x * 64;

  for (int i = tid; i < 64 * 128; i += blockDim.x) {
    int r = i >> 7, c = i & 127;
    sX[r * ST + c] = X[((size_t)b * kENV + row0) * 128 + i];
  }
  for (int i = tid; i < 128 * 128; i += blockDim.x) {
    int r = i >> 7, c = i & 127;
    sW[c * ST + r] = W1[i];
  }
  __syncthreads();

  const int arow = wave * 16, rsel = lane >> 4, rlo = lane & 15;

  v8f acc[8];
#pragma unroll
  for (int nt = 0; nt < 8; ++nt) {
    const float bi = b1[nt * 16 + rlo];
#pragma unroll
    for (int i = 0; i < 8; ++i) acc[nt][i] = bi;
  }
#pragma unroll
  for (int k0 = 0; k0 < 128; k0 += 4) {
    const int ka = k0 + (rsel << 1);
    const v2f a = ld2(&sX[(arow + rlo) * ST + ka]);
#pragma unroll
    for (int nt = 0; nt < 8; ++nt) {
      const v2f bb = ld2(&sW[(nt * 16 + rlo) * ST + ka]);
      acc[nt] = wmma4(a, bb, acc[nt]);
    }
  }
#pragma unroll
  for (int nt = 0; nt < 8; ++nt)
#pragma unroll
    for (int i = 0; i < 8; ++i) {
      const float v = fmaxf(acc[nt][i], 0.f);
      const int r = arow + i + 8 * rsel;
      sH[r * ST + nt * 16 + rlo] = v;
      H[((size_t)b * kENV + row0 + r) * 128 + nt * 16 + rlo] = v;
    }
  __syncthreads();

  if (tid < 64) {
    float acc2 = 0.f;
    for (int d = 0; d < 128; ++d) acc2 += sH[tid * ST + d] * wsv[d];
    logits[(size_t)b * kENV + row0 + tid] = acc2;
  }
}

// ---------------------------------------------------------------------------
// Aggregator stage 2: per-batch softmax over logits, pooled = sum alpha*h
// ---------------------------------------------------------------------------
__launch_bounds__(128)
__global__ void agg_pool_kernel(const float* __restrict__ logits,
                                const float* __restrict__ H,
                                float* __restrict__ pooled) {
  __shared__ float sl[512];
  __shared__ float red[128];
  const int b = blockIdx.x, tid = threadIdx.x;

  for (int i = tid; i < 512; i += 128) sl[i] = logits[(size_t)b * 512 + i];
  __syncthreads();

  float m = -3.4e38f;
  for (int i = tid; i < 512; i += 128) m = fmaxf(m, sl[i]);
  red[tid] = m; __syncthreads();
  for (int s = 64; s > 0; s >>= 1) {
    if (tid < s) red[tid] = fmaxf(red[tid], red[tid + s]);
    __syncthreads();
  }
  m = red[0]; __syncthreads();

  float ssum = 0.f;
  for (int i = tid; i < 512; i += 128) { float e = __expf(sl[i] - m); sl[i] = e; ssum += e; }
  red[tid] = ssum; __syncthreads();
  for (int s = 64; s > 0; s >>= 1) {
    if (tid < s) red[tid] += red[tid + s];
    __syncthreads();
  }
  const float inv = 1.f / red[0];
  __syncthreads();

  float acc = 0.f;
  for (int e = 0; e < 512; ++e)
    acc += sl[e] * H[((size_t)b * 512 + e) * 128 + tid];
  pooled[(size_t)b * 128 + tid] = acc * inv;
}

// ---------------------------------------------------------------------------
// Aggregator stage 3: out = (h + pooled) @ W2 + b2
// ---------------------------------------------------------------------------
__launch_bounds__(128)
__global__ void agg_out_kernel(const float* __restrict__ H,
                               const float* __restrict__ pooled,
                               const float* __restrict__ W2,
                               const float* __restrict__ b2,
                               float* __restrict__ Y) {
  constexpr int ST = 130;
  extern __shared__ float smem[];
  float* sX = smem;
  float* sW = sX + 64 * ST;

  const int tid  = threadIdx.x;
  const int lane = tid & 31;
  const int wave = tid >> 5;
  const int b    = blockIdx.y;
  const int row0 = blockIdx.x * 64;

  for (int i = tid; i < 64 * 128; i += blockDim.x) {
    int r = i >> 7, c = i & 127;
    sX[r * ST + c] = H[((size_t)b * kENV + row0) * 128 + i] + pooled[(size_t)b * 128 + c];
  }
  for (int i = tid; i < 128 * 128; i += blockDim.x) {
    int r = i >> 7, c = i & 127;
    sW[c * ST + r] = W2[i];
  }
  __syncthreads();

  const int arow = wave * 16, rsel = lane >> 4, rlo = lane & 15;

  v8f acc[8];
#pragma unroll
  for (int nt = 0; nt < 8; ++nt) {
    const float bi = b2[nt * 16 + rlo];
#pragma unroll
    for (int i = 0; i < 8; ++i) acc[nt][i] = bi;
  }
#pragma unroll
  for (int k0 = 0; k0 < 128; k0 += 4) {
    const int ka = k0 + (rsel << 1);
    const v2f a = ld2(&sX[(arow + rlo) * ST + ka]);
#pragma unroll
    for (int nt = 0; nt < 8; ++nt) {
      const v2f bb = ld2(&sW[(nt * 16 + rlo) * ST + ka]);
      acc[nt] = wmma4(a, bb, acc[nt]);
    }
  }
#pragma unroll
  for (int nt = 0; nt < 8; ++nt)
#pragma unroll
    for (int i = 0; i < 8; ++i)
      Y[((size_t)b * kENV + row0 + arow + i + 8 * rsel) * 128 + nt * 16 + rlo] =
          acc[nt][i];
}

// ---------------------------------------------------------------------------
// Host orchestration
// ---------------------------------------------------------------------------
extern "C" void kernel_launch(void* const* d_in, const int* in_sizes, int n_in,
                              void* d_out, int out_size, void* d_ws, size_t ws_size,
                              hipStream_t stream) {
  (void)in_sizes; (void)n_in; (void)out_size; (void)ws_size;
  const float* X    = (const float*)d_in[0];
  const float* a1W[12]; for (int i = 0; i < 12; ++i) a1W[i] = (const float*)d_in[1 + i];
  const float* a2W[12]; for (int i = 0; i < 12; ++i) a2W[i] = (const float*)d_in[13 + i];
  const float* g1    = (const float*)d_in[25];
  const float* beta1 = (const float*)d_in[26];
  const float* g2    = (const float*)d_in[27];
  const float* beta2 = (const float*)d_in[28];
  const float* agW1  = (const float*)d_in[29];
  const float* agb1  = (const float*)d_in[30];
  const float* agws  = (const float*)d_in[31];
  const float* agW2  = (const float*)d_in[32];
  const float* agb2  = (const float*)d_in[33];
  float* out = (float*)d_out;

  float* w = (float*)d_ws;
  const size_t n64  = (size_t)kB * kENV * 64;
  const size_t n128 = (size_t)kB * kENV * 128;
  float* Q1 = w;  w += n64;
  float* K1 = w;  w += n64;
  float* V1 = w;  w += n64;
  float* X1 = w;  w += n64;
  float* Q2 = w;  w += n128;
  float* K2 = w;  w += n128;
  float* V2 = w;  w += n128;
  float* X2 = w;  w += n128;
  float* Hh = w;  w += n128;
  float* LG = w;  w += (size_t)kB * kENV;
  float* PL = w;  w += (size_t)kB * 128;

  const dim3 blk(128);
  const dim3 grid8(kENV / 64, kB);

  // LDS sizes (bytes) — match kernel constexpr layouts
  const int lds_mlp1  = (64 * 70 + 64 * 70 + 64 * 66) * 4;
  const int lds_mlp2  = (64 * 66 + 128 * 130 + 64 * 130) * 4;
  const int lds_at64  = (64 * 68 * 3 + 64 * 514) * 4;     // ~184 KB
  const int lds_at128 = (64 * 132 * 3 + 64 * 514) * 4;    // ~233 KB
  const int lds_aggh  = (64 * 130 + 128 * 130 + 64 * 130) * 4;
  const int lds_aggo  = (64 * 130 + 128 * 130) * 4;

  hipFuncSetAttribute((const void*)mlp2_kernel<67, 68, 64, 64>,
                      hipFuncAttributeMaxDynamicSharedMemorySize, lds_mlp1);
  hipFuncSetAttribute((const void*)mlp2_kernel<64, 64, 128, 128>,
                      hipFuncAttributeMaxDynamicSharedMemorySize, lds_mlp2);
  hipFuncSetAttribute((const void*)attn_ln_kernel<64>,
                      hipFuncAttributeMaxDynamicSharedMemorySize, lds_at64);
  hipFuncSetAttribute((const void*)attn_ln_kernel<128>,
                      hipFuncAttributeMaxDynamicSharedMemorySize, lds_at128);
  hipFuncSetAttribute((const void*)agg_h_kernel,
                      hipFuncAttributeMaxDynamicSharedMemorySize, lds_aggh);
  hipFuncSetAttribute((const void*)agg_out_kernel,
                      hipFuncAttributeMaxDynamicSharedMemorySize, lds_aggo);

  // ---- Layer 1 Q/K/V ----
  mlp2_kernel<67, 68, 64, 64><<<grid8, blk, lds_mlp1, stream>>>(
      X, a1W[0], a1W[1], a1W[2], a1W[3], Q1);
  mlp2_kernel<67, 68, 64, 64><<<grid8, blk, lds_mlp1, stream>>>(
      X, a1W[4], a1W[5], a1W[6], a1W[7], K1);
  mlp2_kernel<67, 68, 64, 64><<<grid8, blk, lds_mlp1, stream>>>(
      X, a1W[8], a1W[9], a1W[10], a1W[11], V1);

  // ---- Attention 1 + LN ----
  attn_ln_kernel<64><<<grid8, blk, lds_at64, stream>>>(Q1, K1, V1, g1, beta1, X1);

  // ---- Layer 2 Q/K/V ----
  mlp2_kernel<64, 64, 128, 128><<<grid8, blk, lds_mlp2, stream>>>(
      X1, a2W[0], a2W[1], a2W[2], a2W[3], Q2);
  mlp2_kernel<64, 64, 128, 128><<<grid8, blk, lds_mlp2, stream>>>(
      X1, a2W[4], a2W[5], a2W[6], a2W[7], K2);
  mlp2_kernel<64, 64, 128, 128><<<grid8, blk, lds_mlp2, stream>>>(
      X1, a2W[8], a2W[9], a2W[10], a2W[11], V2);

  // ---- Attention 2 + LN ----
  attn_ln_kernel<128><<<grid8, blk, lds_at128, stream>>>(Q2, K2, V2, g2, beta2, X2);

  // ---- Aggregator ----
  agg_h_kernel<<<grid8, blk, lds_aggh, stream>>>(X2, agW1, agb1, agws, Hh, LG);
  agg_pool_kernel<<<dim3(kB), blk, 0, stream>>>(LG, Hh, PL);
  agg_out_kernel<<<grid8, blk, lds_aggo, stream>>>(Hh, PL, agW2, agb2, out);
}